// SimpleNodeAttn_8521215115943
// MI455X (gfx1250) — compile-verified
//
#include <hip/hip_runtime.h>
#include <math.h>

typedef __attribute__((ext_vector_type(2))) float v2f;
typedef __attribute__((ext_vector_type(8))) float v8f;

#define N_NODES 50000
#define DIM     128
#define HEADS   4
#define NEDGE   800000
#define ATTN_SCALE 0.17677669529663687f   // 1/sqrt(32)

// ---------------------------------------------------------------------------
// Workspace init: segment-max to -1e30, denom and agg to 0. Re-run every call.
// ---------------------------------------------------------------------------
__global__ __launch_bounds__(256)
void nk_init_ws(float* __restrict__ mx, float* __restrict__ den,
                float* __restrict__ agg) {
    int i = blockIdx.x * 256 + threadIdx.x;
    if (i < N_NODES * HEADS) { mx[i] = -1e30f; den[i] = 0.0f; }
    if (i < N_NODES * DIM)   { agg[i] = 0.0f; }
}

// ---------------------------------------------------------------------------
// Fused Q/K/V projection: out = X @ W^T + b using V_WMMA_F32_16X16X4_F32.
// One wave computes one 16x16 output tile; K=128 -> 32 WMMA ops.
// A fragment (16x4 f32): lane&15 = M row, lane>>4 picks K pair {0,1}/{2,3}.
// B fragment (4x16 f32): lane&15 = N col (row of W), same K striping.
// C/D fragment: VGPR r holds row r (+8 for lanes 16-31), col = lane&15.
// blockDim = (32,4): 4 waves = 4 adjacent column tiles. grid = (3125, 2, 3).
// ---------------------------------------------------------------------------
__global__ __launch_bounds__(128)
void nk_qkv_gemm(const float* __restrict__ X,
                 const float* __restrict__ Wq, const float* __restrict__ bq,
                 const float* __restrict__ Wk, const float* __restrict__ bk,
                 const float* __restrict__ Wv, const float* __restrict__ bv,
                 float* __restrict__ Q, float* __restrict__ Kp,
                 float* __restrict__ Vp) {
    const float* W; const float* b; float* out;
    if      (blockIdx.z == 0) { W = Wq; b = bq; out = Q;  }
    else if (blockIdx.z == 1) { W = Wk; b = bk; out = Kp; }
    else                      { W = Wv; b = bv; out = Vp; }

    const int lane = threadIdx.x;                       // 0..31 (wave32)
    const int row0 = blockIdx.x * 16;
    const int col0 = (blockIdx.y * 4 + threadIdx.y) * 16;
    const int m    = lane & 15;
    const int koff = (lane >> 4) * 2;

    const float bias = b[col0 + m];
    v8f acc;
#pragma unroll
    for (int r = 0; r < 8; ++r) acc[r] = bias;

    const float* xrow = X + (row0 + m) * DIM + koff;    // A: rows of X
    const float* wrow = W + (col0 + m) * DIM + koff;    // B: rows of W (== W^T cols)

#pragma unroll 8
    for (int k = 0; k < DIM; k += 4) {
        float2 af = *reinterpret_cast<const float2*>(xrow + k);
        float2 bf = *reinterpret_cast<const float2*>(wrow + k);
        v2f a; a.x = af.x; a.y = af.y;
        v2f bvv; bvv.x = bf.x; bvv.y = bf.y;
        acc = __builtin_amdgcn_wmma_f32_16x16x4_f32(
            false, a, false, bvv, (short)0, acc, false, false);
    }

    const int rbase = (lane >> 4) * 8;
#pragma unroll
    for (int r = 0; r < 8; ++r)
        out[(row0 + rbase + r) * DIM + col0 + m] = acc[r];
}

// ---------------------------------------------------------------------------
// Edge attention logits + segment max. One wave per edge: lane holds 4 dims
// (float4, 16B aligned), 8 lanes per head, shuffle-reduce, atomic f32 max.
// ---------------------------------------------------------------------------
__global__ __launch_bounds__(256)
void nk_edge_scores(const float* __restrict__ Q, const float* __restrict__ K,
                    const int* __restrict__ ei, float* __restrict__ sc,
                    float* __restrict__ mx) {
    const int e    = blockIdx.x * 8 + (threadIdx.x >> 5);
    const int lane = threadIdx.x & 31;
    const int src  = ei[e];
    const int tgt  = ei[NEDGE + e];

    const float4 qv = *reinterpret_cast<const float4*>(Q + tgt * DIM + lane * 4);
    const float4 kv = *reinterpret_cast<const float4*>(K + src * DIM + lane * 4);
    float s = qv.x * kv.x + qv.y * kv.y + qv.z * kv.z + qv.w * kv.w;
    // reduce 8 lanes (32 dims) per head
    s += __shfl_xor(s, 1);
    s += __shfl_xor(s, 2);
    s += __shfl_xor(s, 4);
    if ((lane & 7) == 0) {
        const int h = lane >> 3;
        const float v = s * ATTN_SCALE;
        sc[e * HEADS + h] = v;
        atomicMax(&mx[tgt * HEADS + h], v);   // global_atomic_max_num_f32
    }
}

// ---------------------------------------------------------------------------
// exp(score - mx[tgt]) and segment-sum denominator. Thread per (edge, head).
// ---------------------------------------------------------------------------
__global__ __launch_bounds__(256)
void nk_edge_exp(const int* __restrict__ ei, float* __restrict__ sc,
                 const float* __restrict__ mx, float* __restrict__ den) {
    const int idx = blockIdx.x * 256 + threadIdx.x;     // < E*H
    const int e = idx >> 2;
    const int h = idx & 3;
    const int tgt = ei[NEDGE + e];
    const float ex = __expf(sc[idx] - mx[tgt * HEADS + h]);
    sc[idx] = ex;
    atomicAdd(&den[tgt * HEADS + h], ex);
}

// ---------------------------------------------------------------------------
// Weighted scatter-aggregate: agg[tgt] += V[src] * attn. Wave per edge,
// lane = 4 dims, 4 scalar f32 atomic adds per lane (L2-resident).
// ---------------------------------------------------------------------------
__global__ __launch_bounds__(256)
void nk_edge_agg(const float* __restrict__ V, const int* __restrict__ ei,
                 const float* __restrict__ sc, const float* __restrict__ den,
                 float* __restrict__ agg) {
    const int e    = blockIdx.x * 8 + (threadIdx.x >> 5);
    const int lane = threadIdx.x & 31;
    const int src  = ei[e];
    const int tgt  = ei[NEDGE + e];
    const int h    = lane >> 3;

    const float a = sc[e * HEADS + h] / (den[tgt * HEADS + h] + 1e-10f);
    const float4 vv = *reinterpret_cast<const float4*>(V + src * DIM + lane * 4);
    float* dst = agg + tgt * DIM + lane * 4;
    atomicAdd(dst + 0, vv.x * a);
    atomicAdd(dst + 1, vv.y * a);
    atomicAdd(dst + 2, vv.z * a);
    atomicAdd(dst + 3, vv.w * a);
}

// ---------------------------------------------------------------------------
// Output projection + residual: Y = X + agg @ Wo^T + bo. Same WMMA tiling.
// grid = (3125, 2), block = (32, 4).
// ---------------------------------------------------------------------------
__global__ __launch_bounds__(128)
void nk_out_gemm(const float* __restrict__ AGG, const float* __restrict__ X,
                 const float* __restrict__ Wo, const float* __restrict__ bo,
                 float* __restrict__ Y) {
    const int lane = threadIdx.x;
    const int row0 = blockIdx.x * 16;
    const int col0 = (blockIdx.y * 4 + threadIdx.y) * 16;
    const int m    = lane & 15;
    const int koff = (lane >> 4) * 2;

    const float bias = bo[col0 + m];
    v8f acc;
#pragma unroll
    for (int r = 0; r < 8; ++r) acc[r] = bias;

    const float* arow = AGG + (row0 + m) * DIM + koff;
    const float* wrow = Wo + (col0 + m) * DIM + koff;

#pragma unroll 8
    for (int k = 0; k < DIM; k += 4) {
        float2 af = *reinterpret_cast<const float2*>(arow + k);
        float2 bf = *reinterpret_cast<const float2*>(wrow + k);
        v2f a; a.x = af.x; a.y = af.y;
        v2f bvv; bvv.x = bf.x; bvv.y = bf.y;
        acc = __builtin_amdgcn_wmma_f32_16x16x4_f32(
            false, a, false, bvv, (short)0, acc, false, false);
    }

    const int rbase = (lane >> 4) * 8;
#pragma unroll
    for (int r = 0; r < 8; ++r) {
        const int row = row0 + rbase + r;
        Y[row * DIM + col0 + m] = acc[r] + X[row * DIM + col0 + m];
    }
}

// ---------------------------------------------------------------------------
// In-place LayerNorm over Y rows. Wave per node, full wave32 shuffle reduce.
// ---------------------------------------------------------------------------
__global__ __launch_bounds__(256)
void nk_layernorm(float* __restrict__ Y, const float* __restrict__ g,
                  const float* __restrict__ b) {
    const int n    = blockIdx.x * 8 + (threadIdx.x >> 5);
    const int lane = threadIdx.x & 31;

    float4 y = *reinterpret_cast<const float4*>(Y + n * DIM + lane * 4);
    float s  = y.x + y.y + y.z + y.w;
    float ss = y.x * y.x + y.y * y.y + y.z * y.z + y.w * y.w;
#pragma unroll
    for (int w = 1; w < 32; w <<= 1) {
        s  += __shfl_xor(s, w);
        ss += __shfl_xor(ss, w);
    }
    const float mu  = s * (1.0f / DIM);
    const float var = ss * (1.0f / DIM) - mu * mu;
    const float inv = rsqrtf(var + 1e-5f);

    const float4 gg = *reinterpret_cast<const float4*>(g + lane * 4);
    const float4 bb = *reinterpret_cast<const float4*>(b + lane * 4);
    float4 o;
    o.x = gg.x * (y.x - mu) * inv + bb.x;
    o.y = gg.y * (y.y - mu) * inv + bb.y;
    o.z = gg.z * (y.z - mu) * inv + bb.z;
    o.w = gg.w * (y.w - mu) * inv + bb.w;
    *reinterpret_cast<float4*>(Y + n * DIM + lane * 4) = o;
}

// ---------------------------------------------------------------------------
extern "C" void kernel_launch(void* const* d_in, const int* in_sizes, int n_in,
                              void* d_out, int out_size, void* d_ws, size_t ws_size,
                              hipStream_t stream) {
    const float* X    = (const float*)d_in[0];   // node_features [N, D]
    const float* Wq   = (const float*)d_in[1];
    const float* bq   = (const float*)d_in[2];
    const float* Wk   = (const float*)d_in[3];
    const float* bk   = (const float*)d_in[4];
    const float* Wv   = (const float*)d_in[5];
    const float* bv   = (const float*)d_in[6];
    const float* Wo   = (const float*)d_in[7];
    const float* bo   = (const float*)d_in[8];
    const float* ln_g = (const float*)d_in[9];
    const float* ln_b = (const float*)d_in[10];
    const int*   ei   = (const int*)d_in[11];    // edge_index [2, E]

    float* Y = (float*)d_out;                    // [N, D]

    // Workspace partition (floats)
    const size_t ND = (size_t)N_NODES * DIM;     // 6.4M
    const size_t EH = (size_t)NEDGE * HEADS;     // 3.2M
    const size_t NH = (size_t)N_NODES * HEADS;   // 0.2M
    float* ws  = (float*)d_ws;
    float* Q   = ws;
    float* K   = Q + ND;
    float* V   = K + ND;
    float* sc  = V + ND;                         // edge scores / exp  [E, H]
    float* mx  = sc + EH;                        // segment max        [N, H]
    float* den = mx + NH;                        // segment sum        [N, H]
    float* agg = den + NH;                       // aggregated values  [N, D]

    // 1) init workspace accumulators (deterministic per call)
    nk_init_ws<<<(ND + 255) / 256, 256, 0, stream>>>(mx, den, agg);

    // 2) Q/K/V projections (WMMA f32 16x16x4). 3125 row tiles, 8 col tiles.
    nk_qkv_gemm<<<dim3(N_NODES / 16, 2, 3), dim3(32, 4), 0, stream>>>(
        X, Wq, bq, Wk, bk, Wv, bv, Q, K, V);

    // 3) edge logits + segment max (wave per edge)
    nk_edge_scores<<<NEDGE / 8, 256, 0, stream>>>(Q, K, ei, sc, mx);

    // 4) exp + denom
    nk_edge_exp<<<(NEDGE * HEADS) / 256, 256, 0, stream>>>(ei, sc, mx, den);

    // 5) weighted scatter aggregation
    nk_edge_agg<<<NEDGE / 8, 256, 0, stream>>>(V, ei, sc, den, agg);

    // 6) output projection + residual (WMMA)
    nk_out_gemm<<<dim3(N_NODES / 16, 2), dim3(32, 4), 0, stream>>>(
        agg, X, Wo, bo, Y);

    // 7) LayerNorm in place on Y
    nk_layernorm<<<N_NODES / 8, 256, 0, stream>>>(Y, ln_g, ln_b);
}